// Net3DLayer_58188216926998
// MI455X (gfx1250) — compile-verified
//
#include <hip/hip_runtime.h>
#include <hip/hip_bf16.h>

// ---------------------------------------------------------------------------
// Net3DLayer for MI455X (gfx1250): fp32 WMMA (V_WMMA_F32_16X16X4_F32) GEMMs,
// LDS-staged weights in B-fragment order, atomic-f32 segment sum.
// ---------------------------------------------------------------------------

#define NN 20000
#define EE 320000
#define HH 128
#define K1 384            // 2H + ED
#define XP 388            // x-tile LDS pitch (bank-conflict-free A frags)
#define HP 132            // h/m-tile LDS pitch

typedef float v2f __attribute__((ext_vector_type(2)));
typedef float v8f __attribute__((ext_vector_type(8)));

// ---- LDS layout (floats) for the edge kernel --------------------------------
#define E_OFF_W1  0        // 96*8*64 = 49152 (W1 swizzled, K=384)
#define E_OFF_W2  49152    // 32*8*64 = 16384 (W2 swizzled, K=128)
#define E_OFF_X   65536    // 16*388  = 6208
#define E_OFF_H   71744    // 16*132  = 2112
#define E_OFF_M   73856    // 16*132  = 2112
#define E_OFF_B1  75968    // 128
#define E_OFF_B2  76096    // 128
#define E_OFF_WS  76224    // 128
#define E_OFF_G   76352    // 16
#define E_OFF_DST 76368    // 16 (ints)
#define EDGE_LDS_FLOATS 76384
#define EDGE_LDS_BYTES  (EDGE_LDS_FLOATS * 4)

// node kernels: Usw 16384 + xs 2112 + bias 128
#define N_OFF_X   16384
#define N_OFF_B   18496
#define NODE_LDS_FLOATS 18624
#define NODE_LDS_BYTES  (NODE_LDS_FLOATS * 4)

// ---------------------------------------------------------------------------
// Stage a K x 128 row-major weight matrix into LDS in WMMA B-fragment order:
// flat index = ((kk*8 + t)*32 + lane)*2 + v  holds  W[kk*4 + (lane>>4)*2 + v][t*16 + (lane&15)]
// so each wave's per-k-step B fragment is one conflict-free ds_load_b64.
// ---------------------------------------------------------------------------
__device__ __forceinline__ void stage_swizzle(float* dstLds, const float* __restrict__ W,
                                              int ksteps, int tid, int nthreads) {
  int total = ksteps * 512;  // ksteps * 8 tiles * 32 lanes * 2
  for (int idx = tid; idx < total; idx += nthreads) {
    int v    = idx & 1;
    int lane = (idx >> 1) & 31;
    int t    = (idx >> 6) & 7;
    int kk   = idx >> 9;
    int row  = kk * 4 + ((lane >> 4) << 1) + v;
    int col  = t * 16 + (lane & 15);
    dstLds[idx] = W[row * HH + col];
  }
}

// D(16x16) += A(16xK from LDS tile) * B(Kx16 from swizzled LDS weights)
__device__ __forceinline__ v8f wmma_tile(const float* xs, int pitch,
                                         const float* Wsw, int ksteps,
                                         int waveT, int lane) {
  v8f c = {};
  int row = lane & 15;
  int hi  = (lane >> 4) << 1;           // 0 or 2
  const float* aBase = xs + row * pitch + hi;
  const v2f*   bBase = (const v2f*)Wsw;
#pragma unroll 4
  for (int kk = 0; kk < ksteps; ++kk) {
    v2f a = *(const v2f*)(aBase + kk * 4);
    v2f b = bBase[(kk * 8 + waveT) * 32 + lane];
    c = __builtin_amdgcn_wmma_f32_16x16x4_f32(false, a, false, b, (short)0, c,
                                              false, false);
  }
  return c;
}

// Scatter a C tile (16 rows x 16 cols, this wave's column slice) into an LDS tile.
__device__ __forceinline__ void store_tile_lds(float* tile, int pitch, v8f c,
                                               int waveT, int lane) {
  int col   = waveT * 16 + (lane & 15);
  int rbase = (lane >> 4) * 8;
#pragma unroll
  for (int r = 0; r < 8; ++r) tile[(rbase + r) * pitch + col] = c[r];
}

// ---------------------------------------------------------------------------
// Kernel 0: zero m_sum (feat region of d_out) and the BN stats scratch.
// ---------------------------------------------------------------------------
__global__ void zero_kernel(float* msum, float* ws) {
  size_t i = (size_t)blockIdx.x * blockDim.x + threadIdx.x;
  if (i < (size_t)NN * HH) msum[i] = 0.0f;
  if (i < 512) ws[i] = 0.0f;
}

// ---------------------------------------------------------------------------
// Edge kernel: per 16-edge tile
//   x = [feat[src] | feat[dst] | d]          (16 x 384, LDS)
//   h = relu(x @ W1 + b1)                    (WMMA, K=384)
//   m = silu(h @ W2 + b2)                    (WMMA, K=128)
//   d_new = d + m                            (store)
//   gate = sigmoid(m @ Ws + bs)              (shuffle reduce)
//   atomicAdd(m_sum[dst], m * gate)          (global_atomic_add_f32)
// ---------------------------------------------------------------------------
__global__ __launch_bounds__(256) void edge_kernel(
    const float* __restrict__ feat, const float* __restrict__ dvec,
    const int* __restrict__ src, const int* __restrict__ dst,
    const float* __restrict__ W1, const float* __restrict__ b1,
    const float* __restrict__ W2, const float* __restrict__ b2,
    const float* __restrict__ Wsg, const float* __restrict__ bsg,
    float* msum, float* dnew) {
  extern __shared__ float smem[];
  float* W1s = smem + E_OFF_W1;
  float* W2s = smem + E_OFF_W2;
  float* xs  = smem + E_OFF_X;
  float* hs  = smem + E_OFF_H;
  float* ms  = smem + E_OFF_M;
  float* b1s = smem + E_OFF_B1;
  float* b2s = smem + E_OFF_B2;
  float* wss = smem + E_OFF_WS;
  float* gts = smem + E_OFF_G;
  int*   dsti = (int*)(smem + E_OFF_DST);

  const int tid  = threadIdx.x;
  const int lane = tid & 31;
  const int wv   = tid >> 5;

  stage_swizzle(W1s, W1, 96, tid, 256);
  stage_swizzle(W2s, W2, 32, tid, 256);
  if (tid < HH) { b1s[tid] = b1[tid]; b2s[tid] = b2[tid]; wss[tid] = Wsg[tid]; }
  const float bs0 = bsg[0];
  __syncthreads();

  const int col   = wv * 16 + (lane & 15);
  const int rbase = (lane >> 4) * 8;

  for (int tile = blockIdx.x; tile < EE / 16; tile += gridDim.x) {
    const int ebase = tile * 16;
    if (tid < 16) dsti[tid] = dst[ebase + tid];

    // stage x tile: 16 rows x 96 float4
    for (int j = tid; j < 16 * 96; j += 256) {
      int row = j / 96;
      int c4  = (j % 96) * 4;
      const float* p;
      if (c4 < 128)      p = feat + (size_t)src[ebase + row] * HH + c4;
      else if (c4 < 256) p = feat + (size_t)dst[ebase + row] * HH + (c4 - 128);
      else               p = dvec + (size_t)(ebase + row) * HH + (c4 - 256);
      float4 v = *(const float4*)p;
      *(float4*)(xs + row * XP + c4) = v;
    }
    __syncthreads();

    // GEMM1 + relu
    v8f c = wmma_tile(xs, XP, W1s, 96, wv, lane);
    float bias = b1s[col];
#pragma unroll
    for (int r = 0; r < 8; ++r) {
      float z = c[r] + bias;
      c[r] = z > 0.0f ? z : 0.0f;
    }
    store_tile_lds(hs, HP, c, wv, lane);
    __syncthreads();

    // GEMM2 + silu
    c = wmma_tile(hs, HP, W2s, 32, wv, lane);
    bias = b2s[col];
#pragma unroll
    for (int r = 0; r < 8; ++r) {
      float z = c[r] + bias;
      c[r] = z / (1.0f + __expf(-z));
    }

    // d_new = d + m  (d already staged in xs cols 256..383), stash m for gate
#pragma unroll
    for (int r = 0; r < 8; ++r) {
      int row = rbase + r;
      float dval = xs[row * XP + 256 + col];
      dnew[(size_t)(ebase + row) * HH + col] = dval + c[r];
      ms[row * HP + col] = c[r];
    }
    __syncthreads();

    // gate: each half-wave reduces one edge row
    {
      int row = wv * 2 + (lane >> 4);
      int l   = lane & 15;
      float p = 0.0f;
#pragma unroll
      for (int j = 0; j < 8; ++j) {
        int cc = l + j * 16;
        p += ms[row * HP + cc] * wss[cc];
      }
      p += __shfl_xor(p, 1);
      p += __shfl_xor(p, 2);
      p += __shfl_xor(p, 4);
      p += __shfl_xor(p, 8);
      if (l == 0) gts[row] = 1.0f / (1.0f + __expf(-(p + bs0)));
    }
    __syncthreads();

    // scatter gated messages
#pragma unroll
    for (int r = 0; r < 8; ++r) {
      int row = rbase + r;
      unsafeAtomicAdd(msum + (size_t)dsti[row] * HH + col, c[r] * gts[row]);
    }
    __syncthreads();  // protect xs/dsti/gts before next tile
  }
}

// ---------------------------------------------------------------------------
// Node kernel 1: u = relu((m_sum + feat) @ U1 + bu1), in-place into m_sum
// region; accumulate per-column sum / sumsq for batchnorm.
// ---------------------------------------------------------------------------
__global__ __launch_bounds__(256) void node1_kernel(
    const float* __restrict__ feat, const float* __restrict__ U1,
    const float* __restrict__ bu1, float* uio, float* stats) {
  extern __shared__ float smem[];
  float* Us = smem;
  float* xs = smem + N_OFF_X;
  float* bb = smem + N_OFF_B;
  const int tid  = threadIdx.x;
  const int lane = tid & 31;
  const int wv   = tid >> 5;

  stage_swizzle(Us, U1, 32, tid, 256);
  if (tid < HH) bb[tid] = bu1[tid];

  const int nbase = blockIdx.x * 16;
  for (int j = tid; j < 16 * 32; j += 256) {
    int row = j >> 5;
    int c4  = (j & 31) * 4;
    const float* up = uio + (size_t)(nbase + row) * HH + c4;
    const float* fp = feat + (size_t)(nbase + row) * HH + c4;
    float4 a = *(const float4*)up;
    float4 f = *(const float4*)fp;
    float4 s;
    s.x = a.x + f.x; s.y = a.y + f.y; s.z = a.z + f.z; s.w = a.w + f.w;
    *(float4*)(xs + row * HP + c4) = s;
  }
  __syncthreads();

  v8f c = wmma_tile(xs, HP, Us, 32, wv, lane);
  const int col   = wv * 16 + (lane & 15);
  const int rbase = (lane >> 4) * 8;
  float bias = bb[col];
  float s1 = 0.0f, s2 = 0.0f;
#pragma unroll
  for (int r = 0; r < 8; ++r) {
    float z = c[r] + bias;
    float u = z > 0.0f ? z : 0.0f;
    uio[(size_t)(nbase + rbase + r) * HH + col] = u;
    s1 += u;
    s2 += u * u;
  }
  s1 += __shfl_xor(s1, 16);
  s2 += __shfl_xor(s2, 16);
  if (lane < 16) {
    unsafeAtomicAdd(stats + col, s1);
    unsafeAtomicAdd(stats + HH + col, s2);
  }
}

// ---------------------------------------------------------------------------
// Stats kernel (1 block, 128 threads): fold batchnorm into per-column scale
// a[k] = gamma*rsqrt(var+eps), and precompute cvec = (beta - mu*a) @ U2 + bu2.
// ws layout (floats): [0,128) sum, [128,256) sumsq, [256,384) scale, [384,512) cvec
// ---------------------------------------------------------------------------
__global__ void stats_kernel(const float* __restrict__ U2,
                             const float* __restrict__ bu2,
                             const float* __restrict__ gamma,
                             const float* __restrict__ beta, float* ws) {
  __shared__ float cs[HH];
  int j = threadIdx.x;
  float invN = 1.0f / (float)NN;
  float mu  = ws[j] * invN;
  float var = ws[HH + j] * invN - mu * mu;
  float s   = rsqrtf(var + 1e-5f) * gamma[j];
  ws[256 + j] = s;
  cs[j] = beta[j] - mu * s;
  __syncthreads();
  float acc = bu2[j];
  for (int k = 0; k < HH; ++k) acc += cs[k] * U2[k * HH + j];
  ws[384 + j] = acc;
}

// ---------------------------------------------------------------------------
// Node kernel 2: feat_out = (u * a) @ U2 + cvec + feat, in-place into u region.
// ---------------------------------------------------------------------------
__global__ __launch_bounds__(256) void node2_kernel(
    const float* __restrict__ feat, const float* __restrict__ U2,
    const float* __restrict__ ws, float* uio) {
  extern __shared__ float smem[];
  float* Us = smem;
  float* xs = smem + N_OFF_X;
  const int tid  = threadIdx.x;
  const int lane = tid & 31;
  const int wv   = tid >> 5;

  stage_swizzle(Us, U2, 32, tid, 256);

  const int nbase = blockIdx.x * 16;
  for (int j = tid; j < 16 * 32; j += 256) {
    int row = j >> 5;
    int c4  = (j & 31) * 4;
    float4 a  = *(const float4*)(uio + (size_t)(nbase + row) * HH + c4);
    float4 sc = *(const float4*)(ws + 256 + c4);
    float4 r;
    r.x = a.x * sc.x; r.y = a.y * sc.y; r.z = a.z * sc.z; r.w = a.w * sc.w;
    *(float4*)(xs + row * HP + c4) = r;
  }
  __syncthreads();

  v8f c = wmma_tile(xs, HP, Us, 32, wv, lane);
  const int col   = wv * 16 + (lane & 15);
  const int rbase = (lane >> 4) * 8;
  float cv = ws[384 + col];
#pragma unroll
  for (int r = 0; r < 8; ++r) {
    size_t row = (size_t)(nbase + rbase + r);
    float f = feat[row * HH + col];
    uio[row * HH + col] = c[r] + cv + f;
  }
}

// ---------------------------------------------------------------------------
extern "C" void kernel_launch(void* const* d_in, const int* in_sizes, int n_in,
                              void* d_out, int out_size, void* d_ws, size_t ws_size,
                              hipStream_t stream) {
  const float* feat  = (const float*)d_in[0];
  const float* dvec  = (const float*)d_in[1];
  const int*   src   = (const int*)d_in[2];
  const int*   dst   = (const int*)d_in[3];
  const float* W1    = (const float*)d_in[4];
  const float* b1    = (const float*)d_in[5];
  const float* W2    = (const float*)d_in[6];
  const float* b2    = (const float*)d_in[7];
  const float* Wsg   = (const float*)d_in[8];
  const float* bsg   = (const float*)d_in[9];
  const float* U1    = (const float*)d_in[10];
  const float* bu1   = (const float*)d_in[11];
  const float* gamma = (const float*)d_in[12];
  const float* beta  = (const float*)d_in[13];
  const float* U2    = (const float*)d_in[14];
  const float* bu2   = (const float*)d_in[15];

  float* out  = (float*)d_out;
  float* msum = out;                           // reused: m_sum -> u -> feat_out
  float* dnew = out + (size_t)NN * HH;         // d_new region
  float* ws   = (float*)d_ws;                  // 512 floats of BN stats

  // Raise dynamic-LDS limits (idempotent; host-side, graph-capture safe).
  (void)hipFuncSetAttribute((const void*)edge_kernel,
                            hipFuncAttributeMaxDynamicSharedMemorySize, EDGE_LDS_BYTES);
  (void)hipFuncSetAttribute((const void*)node1_kernel,
                            hipFuncAttributeMaxDynamicSharedMemorySize, NODE_LDS_BYTES);
  (void)hipFuncSetAttribute((const void*)node2_kernel,
                            hipFuncAttributeMaxDynamicSharedMemorySize, NODE_LDS_BYTES);

  zero_kernel<<<(NN * HH + 255) / 256, 256, 0, stream>>>(msum, ws);
  edge_kernel<<<1250, 256, EDGE_LDS_BYTES, stream>>>(feat, dvec, src, dst, W1, b1,
                                                     W2, b2, Wsg, bsg, msum, dnew);
  node1_kernel<<<NN / 16, 256, NODE_LDS_BYTES, stream>>>(feat, U1, bu1, msum, ws);
  stats_kernel<<<1, 128, 0, stream>>>(U2, bu2, gamma, beta, ws);
  node2_kernel<<<NN / 16, 256, NODE_LDS_BYTES, stream>>>(feat, U2, ws, msum);
}